// Masking_11261404250941
// MI455X (gfx1250) — compile-verified
//
#include <hip/hip_runtime.h>

// Cutout masking: out[b,c,y,x] = 0 if (y,x) inside any of N boxes for sample b, else x[b,c,y,x].
// Pure streaming op: 192 MiB total traffic -> ~8.6 us floor at 23.3 TB/s HBM.
// Strategy: wave32-friendly 64-lane blocks, float4 (B128) loads/stores, scalar (SGPR)
// box parameters, column-mask bits hoisted out of the row loop, global_prefetch for
// the next row while the current row streams.

namespace {
constexpr int kB = 128;   // batch
constexpr int kC = 3;     // channels
constexpr int kL = 256;   // spatial size
constexpr int kN = 4;     // boxes per sample
constexpr int kRowsPerBlock = 4;
constexpr int kLanes = kL / 4;  // 64 threads = 2 wave32, one float4 per lane covers a row
}

__global__ __launch_bounds__(kLanes) void cutout_mask_kernel(
    const float* __restrict__ x,
    const int* __restrict__ Xb, const int* __restrict__ Yb,
    const int* __restrict__ Wb, const int* __restrict__ Hb,
    float* __restrict__ out)
{
    const int blk  = blockIdx.x;                         // 0 .. B*L/kRowsPerBlock - 1
    const int b    = blk / (kL / kRowsPerBlock);         // sample index (uniform)
    const int y0   = (blk % (kL / kRowsPerBlock)) * kRowsPerBlock;
    const int lane = threadIdx.x;
    const int px0  = lane * 4;                           // first pixel this lane owns

    // Box parameters: indexed [n*B + b] -> uniform addresses -> s_load_b32.
    // Column-mask bits (which of this lane's 4 pixels fall in box n's x-range)
    // are invariant across the rows this block handles: compute once.
    int ylo[kN], yhi[kN];
    unsigned colm[kN];
#pragma unroll
    for (int n = 0; n < kN; ++n) {
        const int xn = Xb[n * kB + b];
        const int yn = Yb[n * kB + b];
        const int wn = Wb[n * kB + b];
        const int hn = Hb[n * kB + b];
        ylo[n] = yn;
        yhi[n] = yn + hn;
        const int lo = xn, hi = xn + wn;
        unsigned m = 0u;
#pragma unroll
        for (int i = 0; i < 4; ++i) {
            const int xc = px0 + i;
            m |= ((unsigned)((xc >= lo) & (xc <= hi))) << i;
        }
        colm[n] = m;
    }

    const size_t chanStride = (size_t)kL * kL;                       // 65536 floats
    const size_t base = (size_t)b * kC * chanStride + (size_t)y0 * kL + (size_t)px0;
    const float* __restrict__ src = x + base;
    float* __restrict__ dst = out + base;

#pragma unroll
    for (int r = 0; r < kRowsPerBlock; ++r) {
        // Prefetch next row (all channels) while this row streams (global_prefetch_b8).
        if (r + 1 < kRowsPerBlock) {
#pragma unroll
            for (int c = 0; c < kC; ++c)
                __builtin_prefetch(src + (size_t)c * chanStride + (size_t)(r + 1) * kL, 0, 3);
        }

        // Row mask: uniform y-range test per box selects that box's column bits.
        const int y = y0 + r;
        unsigned m = 0u;
#pragma unroll
        for (int n = 0; n < kN; ++n)
            m |= ((y >= ylo[n]) & (y <= yhi[n])) ? colm[n] : 0u;

#pragma unroll
        for (int c = 0; c < kC; ++c) {
            const size_t off = (size_t)c * chanStride + (size_t)r * kL;
            const float4 v = *(const float4*)(src + off);
            float4 o;
            o.x = (m & 1u) ? 0.0f : v.x;
            o.y = (m & 2u) ? 0.0f : v.y;
            o.z = (m & 4u) ? 0.0f : v.z;
            o.w = (m & 8u) ? 0.0f : v.w;
            *(float4*)(dst + off) = o;
        }
    }
}

extern "C" void kernel_launch(void* const* d_in, const int* in_sizes, int n_in,
                              void* d_out, int out_size, void* d_ws, size_t ws_size,
                              hipStream_t stream) {
    (void)in_sizes; (void)n_in; (void)out_size; (void)d_ws; (void)ws_size;
    const float* x = (const float*)d_in[0];
    const int*   X = (const int*)d_in[1];
    const int*   Y = (const int*)d_in[2];
    const int*   W = (const int*)d_in[3];
    const int*   H = (const int*)d_in[4];
    float* out = (float*)d_out;

    const int grid = (kB * kL) / kRowsPerBlock;  // 8192 blocks x 64 threads (2 waves each)
    cutout_mask_kernel<<<grid, kLanes, 0, stream>>>(x, X, Y, W, H, out);
}